// CRF_16389595202091
// MI455X (gfx1250) — compile-verified
//
#include <hip/hip_runtime.h>
#include <hip/hip_bf16.h>

#define B_   64
#define T_   1024
#define D_   1024
#define U_   64
#define KC   128      // K-chunk staged in LDS per iteration
#define WPAD 132      // padded K-stride for transposed W tile (conflict-free b64 reads)

typedef __attribute__((ext_vector_type(2))) float v2f;
typedef __attribute__((ext_vector_type(8))) float v8f;

// ---------------------------------------------------------------------------
// Kernel 1: pot[b,t,u] = inputs[b,t,:] @ W[:,u] + b[u] + start/end boundary
// biases. fp32 WMMA 16x16x4, one wave computes 16 rows x all 64 cols.
// Block = 256 threads (8 waves) -> 128 rows per block, 512 blocks.
// inputs are a 256 MB single-use stream -> non-temporal loads keep W / pot
// hot in the 192 MB L2.
// ---------------------------------------------------------------------------
__global__ __launch_bounds__(256) void crf_potentials_wmma(
    const float* __restrict__ inp, const unsigned char* __restrict__ mask,
    const float* __restrict__ W, const float* __restrict__ bias,
    const float* __restrict__ left_b, const float* __restrict__ right_b,
    float* __restrict__ pot)
{
    __shared__ float sWt[U_ * WPAD];   // transposed W chunk: [n][k], 33 KB

    const int tid  = threadIdx.x;
    const int wave = tid >> 5;
    const int lane = tid & 31;
    const int ln   = lane & 15;        // N-lane within 16
    const int hi   = lane >> 4;        // half-wave selector (K pairs / M+8)
    const int rowBase = blockIdx.x * 128 + wave * 16;
    const int arow    = rowBase + ln;  // A-matrix row this lane feeds

    // Accumulators start at the bias (column-only dependence).
    v8f acc[4];
    #pragma unroll
    for (int nt = 0; nt < 4; ++nt) {
        float bv = bias[nt * 16 + ln];
        #pragma unroll
        for (int e = 0; e < 8; ++e) acc[nt][e] = bv;
    }

    for (int kc = 0; kc < D_ / KC; ++kc) {
        const int k0 = kc * KC;
        __syncthreads();               // protect sWt from previous readers
        // Stage W[k0:k0+KC, 0:64] transposed into LDS (coalesced b128 reads).
        #pragma unroll
        for (int rr = 0; rr < 8; ++rr) {
            int flat = (rr * 256 + tid) * 4;        // element index in chunk
            int kr   = flat >> 6;                   // 0..127
            int n0   = flat & 63;                   // multiple of 4
            const float4 wv = *(const float4*)(W + (size_t)(k0 + kr) * U_ + n0);
            sWt[(n0 + 0) * WPAD + kr] = wv.x;
            sWt[(n0 + 1) * WPAD + kr] = wv.y;
            sWt[(n0 + 2) * WPAD + kr] = wv.z;
            sWt[(n0 + 3) * WPAD + kr] = wv.w;
        }
        __syncthreads();

        const float* aptr = inp + (size_t)arow * D_ + k0 + 2 * hi;
        #pragma unroll 4
        for (int kk = 0; kk < KC; kk += 4) {
            // A fragment 16x4: lanes 0-15 K={kk,kk+1}, lanes 16-31 K={kk+2,kk+3}
            // Single-use stream -> TH_LOAD_NT.
            v2f a = __builtin_nontemporal_load((const v2f*)(aptr + kk));
            #pragma unroll
            for (int nt = 0; nt < 4; ++nt) {
                // B fragment 4x16 from transposed LDS: one 8-byte read
                v2f bf = *(const v2f*)&sWt[(nt * 16 + ln) * WPAD + kk + 2 * hi];
                acc[nt] = __builtin_amdgcn_wmma_f32_16x16x4_f32(
                    /*neg_a=*/false, a, /*neg_b=*/false, bf,
                    /*c_mod=*/(short)0, acc[nt],
                    /*reuse_a=*/false, /*reuse_b=*/false);
            }
        }
    }

    // Epilogue: boundary biases + store. C layout: VGPR e -> M = hi*8 + e,
    // N = nt*16 + ln. pot is re-read by the Viterbi kernel -> default (RT)
    // stores so it stays L2-resident.
    float lb[4], rb[4];
    #pragma unroll
    for (int nt = 0; nt < 4; ++nt) {
        lb[nt] = left_b[nt * 16 + ln];
        rb[nt] = right_b[nt * 16 + ln];
    }
    #pragma unroll
    for (int e = 0; e < 8; ++e) {
        int r = rowBase + hi * 8 + e;          // global row = b*T + t
        int t = r & (T_ - 1);
        bool m  = mask[r] != 0;
        bool mp = (t > 0)      ? (mask[r - 1] != 0) : false;  // per-batch pad
        bool mn = (t < T_ - 1) ? (mask[r + 1] != 0) : false;
        float sl = (m && !mp) ? 1.0f : 0.0f;   // start_mask
        float sr = (m && !mn) ? 1.0f : 0.0f;   // end_mask
        #pragma unroll
        for (int nt = 0; nt < 4; ++nt) {
            int n = nt * 16 + ln;
            pot[(size_t)r * U_ + n] = acc[nt][e] + sl * lb[nt] + sr * rb[nt];
        }
    }
}

// ---------------------------------------------------------------------------
// Kernel 2: fused Viterbi forward scan + backtrack + one-hot write.
// One block per batch. 256 threads = 64 tags (j) x 4 reduction groups (g).
// Backpointers live entirely in LDS (64 KB) so the sequential backtrack is
// a chain of LDS broadcasts, not dependent global loads.
// ---------------------------------------------------------------------------
__global__ __launch_bounds__(256) void crf_viterbi(
    const float* __restrict__ pot, const unsigned char* __restrict__ mask,
    const float* __restrict__ trans, float* __restrict__ onehot)
{
    __shared__ unsigned char s_bp[T_ * U_];    // 64 KB backpointers
    __shared__ float s_state[2][U_];
    __shared__ float s_pv[4][U_];
    __shared__ int   s_pi[4][U_];
    __shared__ unsigned char s_mask[T_];

    const int b   = blockIdx.x;
    const int tid = threadIdx.x;
    const int j   = tid & 63;                  // destination tag
    const int g   = tid >> 6;                  // i-range group (16 each)

    for (int q = tid; q < T_; q += 256) s_mask[q] = mask[b * T_ + q];

    // This thread's slice of trans[:, j] stays in registers for all T steps.
    float tc[16];
    #pragma unroll
    for (int kk = 0; kk < 16; ++kk) tc[kk] = trans[(g * 16 + kk) * U_ + j];

    if (g == 0) s_state[0][j] = pot[((size_t)b * T_) * U_ + j];
    __syncthreads();

    int pr = 0;
    for (int t = 1; t < T_; ++t) {
        // Partial (max, first-argmax) over i in [g*16, g*16+16).
        float m = -3.402823466e38f;
        int am = g * 16;
        #pragma unroll
        for (int kk = 0; kk < 16; ++kk) {
            float sc = s_state[pr][g * 16 + kk] + tc[kk];
            if (sc > m) { m = sc; am = g * 16 + kk; }
        }
        s_pv[g][j] = m;
        s_pi[g][j] = am;
        __syncthreads();
        if (g == 0) {
            // Combine ascending g with strict '>' => first-occurrence argmax.
            float mm = s_pv[0][j]; int aa = s_pi[0][j];
            #pragma unroll
            for (int q = 1; q < 4; ++q) {
                float v = s_pv[q][j];
                if (v > mm) { mm = v; aa = s_pi[q][j]; }
            }
            bool mk = s_mask[t] != 0;
            // pot row is read exactly once here -> non-temporal.
            float newv = __builtin_nontemporal_load(
                             pot + ((size_t)b * T_ + t) * U_ + j) + mm;
            s_state[pr ^ 1][j] = mk ? newv : s_state[pr][j];
            s_bp[t * U_ + j]   = (unsigned char)(mk ? aa : j);
        }
        __syncthreads();
        pr ^= 1;
    }

    // last_tag = first argmax of final state (redundant uniform scan,
    // LDS broadcast reads, no extra barrier needed).
    float mm = s_state[pr][0];
    int tag = 0;
    #pragma unroll 8
    for (int q = 1; q < U_; ++q) {
        float v = s_state[pr][q];
        if (v > mm) { mm = v; tag = q; }
    }

    // Backtrack through LDS backpointers, writing one-hot rows as we go.
    // onehot is write-once, never re-read on device -> non-temporal stores.
    for (int t = T_ - 1; t >= 1; --t) {
        int wt = s_mask[t] ? tag : 0;          // output tag is masked
        if (tid < U_)
            __builtin_nontemporal_store((tid == wt) ? 1.0f : 0.0f,
                onehot + ((size_t)b * T_ + t) * U_ + tid);
        tag = s_bp[t * U_ + tag];              // uniform broadcast read
    }
    int wt0 = s_mask[0] ? tag : 0;
    if (tid < U_)
        __builtin_nontemporal_store((tid == wt0) ? 1.0f : 0.0f,
            onehot + ((size_t)b * T_) * U_ + tid);
}

// ---------------------------------------------------------------------------
extern "C" void kernel_launch(void* const* d_in, const int* in_sizes, int n_in,
                              void* d_out, int out_size, void* d_ws, size_t ws_size,
                              hipStream_t stream)
{
    (void)in_sizes; (void)n_in; (void)out_size; (void)d_ws; (void)ws_size;
    const float*         inp     = (const float*)d_in[0];
    const unsigned char* mask    = (const unsigned char*)d_in[1];
    const float*         W       = (const float*)d_in[2];
    const float*         bias    = (const float*)d_in[3];
    const float*         trans   = (const float*)d_in[4];
    const float*         left_b  = (const float*)d_in[5];
    const float*         right_b = (const float*)d_in[6];

    float* pot    = (float*)d_out;                       // output 0: (B,T,U)
    float* onehot = pot + (size_t)B_ * T_ * U_;          // output 1: (B,T,U)

    crf_potentials_wmma<<<(B_ * T_) / 128, 256, 0, stream>>>(
        inp, mask, W, bias, left_b, right_b, pot);
    crf_viterbi<<<B_, 256, 0, stream>>>(pot, mask, trans, onehot);
}